// GCN_88227218195278
// MI455X (gfx1250) — compile-verified
//
#include <hip/hip_runtime.h>

typedef __attribute__((ext_vector_type(2))) float v2f;
typedef __attribute__((ext_vector_type(8))) float v8f;

#define IN_CH 512
#define HID   32
#define OUT_CH 64

// ---------------------------------------------------------------------------
// Degree / normalization kernels
// ---------------------------------------------------------------------------
__global__ void init_deg_kernel(float* __restrict__ deg, int n) {
    int i = blockIdx.x * blockDim.x + threadIdx.x;
    if (i < n) deg[i] = 1.0f;  // self-loop contributes 1 to degree
}

__global__ void count_deg_kernel(const long long* __restrict__ dst,
                                 float* __restrict__ deg, long long nEdges) {
    long long e = (long long)blockIdx.x * blockDim.x + threadIdx.x;
    if (e < nEdges) atomicAdd(&deg[dst[e]], 1.0f);
}

__global__ void finalize_dinv_kernel(float* __restrict__ deg, int n) {
    int i = blockIdx.x * blockDim.x + threadIdx.x;
    if (i < n) deg[i] = rsqrtf(deg[i]);  // deg >= 1 always (self loop)
}

__global__ void compute_norm_kernel(const long long* __restrict__ src,
                                    const long long* __restrict__ dst,
                                    const float* __restrict__ dinv,
                                    float* __restrict__ normE, long long nEdges) {
    long long e = (long long)blockIdx.x * blockDim.x + threadIdx.x;
    if (e < nEdges) normE[e] = dinv[src[e]] * dinv[dst[e]];
}

// ---------------------------------------------------------------------------
// WMMA f32 GEMM: out[N,F] = (RELU_IN ? relu(in) : in)[N,K] @ W[K,F]
// One wave (32 threads) computes a full 16-row x F output stripe:
// F/16 accumulators share a single A load per k-step, so the big A matrix
// (x = 205 MB > 192 MB L2 for layer 1) is streamed from HBM exactly once.
// A-layout (16x4 f32): lanes 0-15 hold M=lane, K=0/1 in v[0]/v[1];
//                      lanes 16-31 hold M=lane-16, K=2/3.
// B-layout (4x16 f32): lanes 0-15 N=lane, K=0/1; lanes 16-31 K=2/3.
// C/D: VGPR i -> row (i + 8*(lane>=16)), col (lane&15).
// ---------------------------------------------------------------------------
template <int K, int F, bool RELU_IN>
__global__ __launch_bounds__(32)
void gemm_wmma_f32(const float* __restrict__ in, const float* __restrict__ W,
                   float* __restrict__ out) {
    constexpr int NT = F / 16;          // column tiles per wave
    const int lane  = threadIdx.x;
    const int half  = lane >> 4;        // 0 or 1
    const int l16   = lane & 15;
    const int tileM = blockIdx.x;

    const float* __restrict__ arow = in + (size_t)(tileM * 16 + l16) * K + 2 * half;
    const float* __restrict__ bcol = W + (size_t)(2 * half) * F + l16;

    v8f c[NT];
#pragma unroll
    for (int t = 0; t < NT; ++t) c[t] = (v8f){};

#pragma unroll 2
    for (int k = 0; k < K; k += 4) {
        float a0 = arow[k + 0];
        float a1 = arow[k + 1];
        if (RELU_IN) { a0 = fmaxf(a0, 0.0f); a1 = fmaxf(a1, 0.0f); }
        v2f a; a.x = a0; a.y = a1;
#pragma unroll
        for (int t = 0; t < NT; ++t) {
            v2f b;
            b.x = bcol[(size_t)k * F + t * 16];
            b.y = bcol[(size_t)(k + 1) * F + t * 16];
            c[t] = __builtin_amdgcn_wmma_f32_16x16x4_f32(
                       /*neg_a=*/false, a, /*neg_b=*/false, b,
                       /*c_mod=*/(short)0, c[t], /*reuse_a=*/false, /*reuse_b=*/false);
        }
    }

    float* __restrict__ orow =
        out + (size_t)(tileM * 16 + 8 * half) * F + l16;
#pragma unroll
    for (int t = 0; t < NT; ++t)
#pragma unroll
        for (int i = 0; i < 8; ++i) orow[(size_t)i * F + t * 16] = c[t][i];
}

// ---------------------------------------------------------------------------
// agg[i,f] = bias[f] + dinv[i]^2 * h[i,f]   (self-loop term + bias, also init)
// ---------------------------------------------------------------------------
template <int F>
__global__ void agg_init_kernel(const float* __restrict__ h,
                                const float* __restrict__ dinv,
                                const float* __restrict__ bias,
                                float* __restrict__ agg, int n) {
    long long t = (long long)blockIdx.x * blockDim.x + threadIdx.x;
    long long i = t / F;
    int f = (int)(t % F);
    if (i >= n) return;
    float di = dinv[i];
    agg[t] = bias[f] + di * di * h[t];
}

// ---------------------------------------------------------------------------
// Edge scatter: agg[dst,f] += normE[e] * h[src,f]
// Thread per (edge, feature); one wave covers one edge when F==32 (coalesced).
// ---------------------------------------------------------------------------
template <int F>
__global__ void edge_scatter_kernel(const long long* __restrict__ src,
                                    const long long* __restrict__ dst,
                                    const float* __restrict__ normE,
                                    const float* __restrict__ h,
                                    float* __restrict__ agg, long long nEdges) {
    long long t = (long long)blockIdx.x * blockDim.x + threadIdx.x;
    long long e = t / F;
    int f = (int)(t % F);
    if (e >= nEdges) return;
    long long s = src[e];
    long long d = dst[e];
    float v = normE[e] * h[s * F + f];
    atomicAdd(&agg[d * F + f], v);
}

// ---------------------------------------------------------------------------
// Row-wise log-softmax over 64 channels. One wave per row, 8 rows per block.
// ---------------------------------------------------------------------------
__global__ __launch_bounds__(256)
void log_softmax64_kernel(const float* __restrict__ agg,
                          float* __restrict__ out, int n) {
    int row  = blockIdx.x * 8 + (threadIdx.x >> 5);
    int lane = threadIdx.x & 31;
    if (row >= n) return;
    const float* a = agg + (size_t)row * 64;
    float v0 = a[lane];
    float v1 = a[lane + 32];
    float m = fmaxf(v0, v1);
#pragma unroll
    for (int off = 16; off > 0; off >>= 1)
        m = fmaxf(m, __shfl_xor(m, off, 32));
    float s = __expf(v0 - m) + __expf(v1 - m);
#pragma unroll
    for (int off = 16; off > 0; off >>= 1)
        s += __shfl_xor(s, off, 32);
    float lse = m + __logf(s);
    out[(size_t)row * 64 + lane] = v0 - lse;
    out[(size_t)row * 64 + lane + 32] = v1 - lse;
}

// ---------------------------------------------------------------------------
// Host-side launch
// ---------------------------------------------------------------------------
extern "C" void kernel_launch(void* const* d_in, const int* in_sizes, int n_in,
                              void* d_out, int out_size, void* d_ws, size_t ws_size,
                              hipStream_t stream) {
    const float* x      = (const float*)d_in[0];
    const long long* ei = (const long long*)d_in[1];
    const float* W1 = (const float*)d_in[2];
    const float* b1 = (const float*)d_in[3];
    const float* W2 = (const float*)d_in[4];
    const float* b2 = (const float*)d_in[5];
    const float* W3 = (const float*)d_in[6];
    const float* b3 = (const float*)d_in[7];
    float* out = (float*)d_out;

    const int N       = in_sizes[0] / IN_CH;   // 100000
    const long long E = in_sizes[1] / 2;       // 1600000
    const long long* src = ei;                 // edge_index[0]
    const long long* dst = ei + E;             // edge_index[1]

    // Workspace layout (floats): dinv[N] | normE[E] | bufA[N*64] | bufB[N*64]
    float* dinv  = (float*)d_ws;
    float* normE = dinv + N;
    float* bufA  = normE + E;
    float* bufB  = bufA + (size_t)N * 64;

    const int TB = 256;
    dim3 blk(TB);

    // --- normalization: deg -> dinv -> per-edge norm ---
    init_deg_kernel<<<(N + TB - 1) / TB, blk, 0, stream>>>(dinv, N);
    count_deg_kernel<<<(unsigned)((E + TB - 1) / TB), blk, 0, stream>>>(dst, dinv, E);
    finalize_dinv_kernel<<<(N + TB - 1) / TB, blk, 0, stream>>>(dinv, N);
    compute_norm_kernel<<<(unsigned)((E + TB - 1) / TB), blk, 0, stream>>>(src, dst, dinv, normE, E);

    // --- Layer 1: h1 = x @ W1 ; agg1 = scatter(norm*h1) + b1 ---
    gemm_wmma_f32<IN_CH, HID, false>
        <<<dim3(N / 16), dim3(32), 0, stream>>>(x, W1, bufA);
    agg_init_kernel<HID>
        <<<(unsigned)(((long long)N * HID + TB - 1) / TB), blk, 0, stream>>>(bufA, dinv, b1, bufB, N);
    edge_scatter_kernel<HID>
        <<<(unsigned)((E * HID + TB - 1) / TB), blk, 0, stream>>>(src, dst, normE, bufA, bufB, E);

    // --- Layer 2: h2 = relu(agg1) @ W2 ; agg2 ---
    gemm_wmma_f32<HID, HID, true>
        <<<dim3(N / 16), dim3(32), 0, stream>>>(bufB, W2, bufA);
    agg_init_kernel<HID>
        <<<(unsigned)(((long long)N * HID + TB - 1) / TB), blk, 0, stream>>>(bufA, dinv, b2, bufB, N);
    edge_scatter_kernel<HID>
        <<<(unsigned)((E * HID + TB - 1) / TB), blk, 0, stream>>>(src, dst, normE, bufA, bufB, E);

    // --- Layer 3: h3 = relu(agg2) @ W3 ; agg3 ---
    gemm_wmma_f32<HID, OUT_CH, true>
        <<<dim3(N / 16), dim3(32), 0, stream>>>(bufB, W3, bufA);
    agg_init_kernel<OUT_CH>
        <<<(unsigned)(((long long)N * OUT_CH + TB - 1) / TB), blk, 0, stream>>>(bufA, dinv, b3, bufB, N);
    edge_scatter_kernel<OUT_CH>
        <<<(unsigned)((E * OUT_CH + TB - 1) / TB), blk, 0, stream>>>(src, dst, normE, bufA, bufB, E);

    // --- log-softmax ---
    log_softmax64_kernel<<<(N + 7) / 8, dim3(256), 0, stream>>>(bufB, out, N);
}